// SAGEConv_7945689498280
// MI455X (gfx1250) — compile-verified
//
#include <hip/hip_runtime.h>
#include <hip/hip_bf16.h>

// ---------------------------------------------------------------------------
// SAGEConv: out = feat @ w_self + (segment_mean(feat[src], dst)) @ w_neigh + b
// N=100000, E=1600000, D=128, fp32.
//
// Phase 1: zero msg_sum/deg workspace.
// Phase 2: edge scatter: 32 lanes/edge, float4 gather of feat[src] (L2-hot)
//          + 4x global_atomic_add_f32 into msg_sum[dst]; lane0 bumps degree.
// Phase 3: fused mean + dual GEMM with V_WMMA_F32_16X16X4_F32.
//          A-tiles (feat row tile + msg row tile) staged ONCE per block into
//          LDS via GLOBAL_LOAD_ASYNC_TO_LDS_B128 (ASYNCcnt-tracked), removing
//          the 8x redundant per-wave global reads; bank-conflict-free padding.
// ---------------------------------------------------------------------------

typedef float v2f __attribute__((ext_vector_type(2)));
typedef float v8f __attribute__((ext_vector_type(8)));
typedef int v4i __attribute__((ext_vector_type(4)));

#define D_FEAT 128
#define LDS_STRIDE 132  // 132 mod 64 == 4 -> conflict-free float2 A-fragment reads

#if defined(__has_builtin)
#if __has_builtin(__builtin_amdgcn_global_load_async_to_lds_b128) && \
    __has_builtin(__builtin_amdgcn_s_wait_asynccnt)
#define SAGE_ASYNC_LDS 1
#endif
#endif

// ---------------- Phase 1: zero workspace (msg_sum + deg) -------------------
__global__ __launch_bounds__(256) void sage_zero_ws(float4* __restrict__ ws,
                                                    int n_vec4) {
  int i = blockIdx.x * blockDim.x + threadIdx.x;
  if (i < n_vec4) {
    ws[i] = make_float4(0.f, 0.f, 0.f, 0.f);
  }
}

// ---------------- Phase 2: edge scatter-add ---------------------------------
__global__ __launch_bounds__(256) void sage_scatter(
    const float* __restrict__ feat, const int* __restrict__ src,
    const int* __restrict__ dst, float* __restrict__ msg,
    float* __restrict__ deg, int n_edges) {
  const int lane = threadIdx.x & 31;
  const int e = blockIdx.x * 8 + (threadIdx.x >> 5);
  if (e >= n_edges) return;
  const int s = src[e];
  const int d = dst[e];
  const float4 v =
      *(const float4*)(feat + (size_t)s * D_FEAT + (size_t)lane * 4);
  float* mp = msg + (size_t)d * D_FEAT + (size_t)lane * 4;
  atomicAdd(mp + 0, v.x);  // global_atomic_add_f32 (no return): L2 RMW
  atomicAdd(mp + 1, v.y);
  atomicAdd(mp + 2, v.z);
  atomicAdd(mp + 3, v.w);
  if (lane == 0) atomicAdd(deg + d, 1.0f);
}

// ---------------- Phase 3: fused mean + dual GEMM (WMMA f32) ----------------
// Block = 256 threads = 8 waves; block owns a 16-node row tile; wave w owns
// output cols [16w,16w+16). A-tiles come from LDS; B slices stream from
// global (disjoint per wave, L2/WGP$-hot). K=128 in steps of 4.
__global__ __launch_bounds__(256) void sage_gemm_wmma(
    const float* __restrict__ feat, const float* __restrict__ msg,
    const float* __restrict__ deg, const float* __restrict__ w_self,
    const float* __restrict__ w_neigh, const float* __restrict__ b_neigh,
    float* __restrict__ out) {
  __shared__ float ldsA[2][16 * LDS_STRIDE];  // [0]=feat tile, [1]=msg tile

  const int lane = threadIdx.x & 31;
  const int wave = threadIdx.x >> 5;
  const int tileM = blockIdx.x;
  const int nsub = lane & 15;  // A row / B,D column within tile
  const int hi = lane >> 4;    // half-wave select
  const int koff = hi << 1;    // A/B fragment holds K = k+koff, k+koff+1
  const int colBase = wave * 16;

  // ---- Stage both 16x128 A-tiles into LDS (1024 x 16B chunks, 4/thread) ---
  const float* __restrict__ gsrc[2] = {feat + (size_t)tileM * 16 * D_FEAT,
                                       msg + (size_t)tileM * 16 * D_FEAT};
#pragma unroll
  for (int i = 0; i < 4; ++i) {
    const int c = threadIdx.x + (i << 8);  // 0..1023
    const int arr = c >> 9;                // 0=feat, 1=msg
    const int cc = c & 511;
    const int row = cc >> 5;               // 0..15
    const int foff = (cc & 31) << 2;       // float offset, multiple of 4
    const float* g = gsrc[arr] + (size_t)row * D_FEAT + foff;
    float* l = &ldsA[arr][row * LDS_STRIDE + foff];
#ifdef SAGE_ASYNC_LDS
    // GLOBAL_LOAD_ASYNC_TO_LDS_B128: 16B/lane direct to LDS, ASYNCcnt-tracked.
    __builtin_amdgcn_global_load_async_to_lds_b128(
        (__attribute__((address_space(1))) v4i*)g,
        (__attribute__((address_space(3))) v4i*)l, 0, 0);
#else
    *(float4*)l = *(const float4*)g;
#endif
  }
#ifdef SAGE_ASYNC_LDS
  __builtin_amdgcn_s_wait_asynccnt(0);
#endif
  __syncthreads();

  const int arow = tileM * 16 + nsub;
  const float dg = deg[arow];
  const float inv = 1.0f / fmaxf(dg, 1.0f);  // segment-mean (0-degree -> /1)

  const float* __restrict__ lf = &ldsA[0][nsub * LDS_STRIDE];
  const float* __restrict__ lm = &ldsA[1][nsub * LDS_STRIDE];

  v8f acc = {};  // 16x16 f32 C/D tile (8 VGPRs)
#pragma unroll 8
  for (int k = 0; k < D_FEAT; k += 4) {
    // A fragments from LDS: ds_load_b64, conflict-free via LDS_STRIDE pad.
    const v2f a_s = *(const v2f*)(lf + k + koff);
    const v2f m2 = *(const v2f*)(lm + k + koff);
    v2f a_n;
    a_n.x = m2.x * inv;
    a_n.y = m2.y * inv;
    // B fragments: rows k+koff, k+koff+1 of row-major 128x128 weights.
    v2f b_s, b_n;
    b_s.x = w_self[(size_t)(k + koff) * D_FEAT + colBase + nsub];
    b_s.y = w_self[(size_t)(k + koff + 1) * D_FEAT + colBase + nsub];
    b_n.x = w_neigh[(size_t)(k + koff) * D_FEAT + colBase + nsub];
    b_n.y = w_neigh[(size_t)(k + koff + 1) * D_FEAT + colBase + nsub];
    acc = __builtin_amdgcn_wmma_f32_16x16x4_f32(
        false, a_s, false, b_s, (short)0, acc, false, false);
    acc = __builtin_amdgcn_wmma_f32_16x16x4_f32(
        false, a_n, false, b_n, (short)0, acc, false, false);
  }

  const float bias = b_neigh[colBase + nsub];
  const int rbase = tileM * 16 + (hi << 3);  // D row base for this half-wave
#pragma unroll
  for (int v = 0; v < 8; ++v) {
    out[(size_t)(rbase + v) * D_FEAT + colBase + nsub] = acc[v] + bias;
  }
}

// ---------------------------------------------------------------------------
extern "C" void kernel_launch(void* const* d_in, const int* in_sizes, int n_in,
                              void* d_out, int out_size, void* d_ws,
                              size_t ws_size, hipStream_t stream) {
  const float* feat = (const float*)d_in[0];     // [N,128]
  const float* w_self = (const float*)d_in[1];   // [128,128]
  const float* w_neigh = (const float*)d_in[2];  // [128,128]
  const float* b_neigh = (const float*)d_in[3];  // [128]
  const int* src = (const int*)d_in[4];          // [E]
  const int* dst = (const int*)d_in[5];          // [E]
  float* out = (float*)d_out;

  const int n_nodes = in_sizes[0] / D_FEAT;
  const int n_edges = in_sizes[4];

  // Workspace: msg_sum [N*128 f32] then deg [N f32].
  float* msg = (float*)d_ws;
  float* deg = msg + (size_t)n_nodes * D_FEAT;

  {  // Phase 1: zero msg+deg (re-runs on every graph replay).
    const int n_f = n_nodes * D_FEAT + n_nodes;
    const int n_v4 = (n_f + 3) / 4;
    const int blocks = (n_v4 + 255) / 256;
    sage_zero_ws<<<blocks, 256, 0, stream>>>((float4*)d_ws, n_v4);
  }

  {  // Phase 2: edge scatter (8 edges per 256-thread block).
    const int blocks = (n_edges + 7) / 8;
    sage_scatter<<<blocks, 256, 0, stream>>>(feat, src, dst, msg, deg,
                                             n_edges);
  }

  {  // Phase 3: fused mean + dual WMMA GEMM (N divisible by 16).
    const int blocks = n_nodes / 16;  // 6250 row tiles
    sage_gemm_wmma<<<blocks, 256, 0, stream>>>(feat, msg, deg, w_self,
                                               w_neigh, b_neigh, out);
  }
}